// CompactPointMLP_3332894621724
// MI455X (gfx1250) — compile-verified
//
#include <hip/hip_runtime.h>
#include <hip/hip_bf16.h>

#define BQ 16
#define NPTS 2048
#define KNN 32
#define MROWS (BQ * NPTS)      // 32768
#define EPSV 1e-5f

typedef __attribute__((ext_vector_type(2))) float v2f;
typedef __attribute__((ext_vector_type(8))) float v8f;

// ---------------------------------------------------------------------------
// Weight transpose + pad:  W[K x Nc] -> Wt[Npad x K] (column-major view),
// zero-filled for c >= Nc. Lets the GEMM read each lane's B pair as one
// contiguous b64 load with no bounds predication.
// ---------------------------------------------------------------------------
__global__ void transpose_pad_w(const float* __restrict__ W,
                                float* __restrict__ Wt, int K, int Nc,
                                int Npad) {
  int i = blockIdx.x * 256 + threadIdx.x;
  if (i >= Npad * K) return;
  int c = i / K, k = i - c * K;
  Wt[i] = (c < Nc) ? W[(size_t)k * Nc + c] : 0.f;
}

// ---------------------------------------------------------------------------
// WMMA f32 GEMM:  Y[M x Nc] = A[M x K] @ W[K x Nc]  (W pre-transposed/padded)
// Requires M % 16 == 0 and K % 4 == 0 (guaranteed by all call sites).
// One wave per 16x16 output tile, V_WMMA_F32_16X16X4_F32 over K in steps of 4.
// Inner loop: 2 unconditional global_load_b64 + 1 wmma; EXEC stays all-ones.
// ---------------------------------------------------------------------------
__global__ __launch_bounds__(256) void gemm_wmma_f32(
    const float* __restrict__ A, const float* __restrict__ Wt,
    float* __restrict__ Y, int K, int Nc, int lda, int ldy, int ntn,
    int tiles) {
  int t = blockIdx.x * 8 + (threadIdx.x >> 5);  // wave-uniform tile id
  if (t >= tiles) return;                       // whole-wave exit
  int tm = t / ntn, tn = t - tm * ntn;
  int lane = threadIdx.x & 31;
  int half = lane >> 4;                         // K pair {0,1} vs {2,3}
  int mn = lane & 15;
  // A: lane holds row mn of the tile, K offsets 2*half + {0,1}
  const float* arow = A + (size_t)(tm * 16 + mn) * lda + 2 * half;
  // B: lane holds column mn of the tile (transposed layout, stride K)
  const float* bcol = Wt + (size_t)(tn * 16 + mn) * K + 2 * half;

  v8f acc = {0.f, 0.f, 0.f, 0.f, 0.f, 0.f, 0.f, 0.f};
  for (int k0 = 0; k0 < K; k0 += 4) {
    v2f a = *(const v2f*)(arow + k0);
    v2f b = *(const v2f*)(bcol + k0);
    // (neg_a, A, neg_b, B, c_mod, C, reuse_a, reuse_b)
    acc = __builtin_amdgcn_wmma_f32_16x16x4_f32(false, a, false, b, (short)0,
                                                acc, false, false);
  }
  int cn = tn * 16 + mn;
  if (cn < Nc) {
    // D layout: lane holds column mn; VGPR v holds row v + 8*half
    float* yout = Y + (size_t)(tm * 16 + 8 * half) * ldy + cn;
#pragma unroll
    for (int v = 0; v < 8; ++v) yout[(size_t)v * ldy] = acc[v];
  }
}

// ---------------------------------------------------------------------------
// Stem: (B,3,N) -> row-major (B*N, 8) raw matmul (K=3, too small for WMMA)
// ---------------------------------------------------------------------------
__global__ void stem_kernel(const float* __restrict__ x,
                            const float* __restrict__ W,
                            float* __restrict__ Y, int ldy) {
  int row = blockIdx.x * 256 + threadIdx.x;
  if (row >= MROWS) return;
  int b = row >> 11, n = row & (NPTS - 1);
  const float* xb = x + (size_t)b * 3 * NPTS;
  float x0 = xb[n], x1 = xb[NPTS + n], x2 = xb[2 * NPTS + n];
  float* y = Y + (size_t)row * ldy;
#pragma unroll
  for (int c = 0; c < 8; ++c)
    y[c] = x0 * W[c] + x1 * W[8 + c] + x2 * W[16 + c];
}

// ---------------------------------------------------------------------------
// BatchNorm stats: one block per channel, deterministic LDS tree reduce.
// stats[2c] = sum, stats[2c+1] = sum of squares over M rows.
// ---------------------------------------------------------------------------
__global__ __launch_bounds__(256) void bn_stats(const float* __restrict__ Y,
                                                int M, int ldy,
                                                float* __restrict__ stats) {
  __shared__ float ssum[256], ssq[256];
  int c = blockIdx.x;
  float s = 0.f, q = 0.f;
  for (int r = threadIdx.x; r < M; r += 256) {
    float v = Y[(size_t)r * ldy + c];
    s += v;
    q += v * v;
  }
  ssum[threadIdx.x] = s;
  ssq[threadIdx.x] = q;
  __syncthreads();
  for (int o = 128; o; o >>= 1) {
    if (threadIdx.x < o) {
      ssum[threadIdx.x] += ssum[threadIdx.x + o];
      ssq[threadIdx.x] += ssq[threadIdx.x + o];
    }
    __syncthreads();
  }
  if (threadIdx.x == 0) {
    stats[2 * c] = ssum[0];
    stats[2 * c + 1] = ssq[0];
  }
}

// ---------------------------------------------------------------------------
// BN apply (+optional residual, +optional relu):  out = act(bn(y) [+ resid])
// ---------------------------------------------------------------------------
__global__ void bn_apply(const float* __restrict__ Y, int ldy,
                         float* __restrict__ O, int ldo,
                         const float* __restrict__ stats,
                         const float* __restrict__ g,
                         const float* __restrict__ b,
                         const float* __restrict__ resid, int ldr,
                         int M, int C, int relu) {
  int i = blockIdx.x * 256 + threadIdx.x;
  if (i >= M * C) return;
  int r = i / C, c = i - r * C;
  float mean = stats[2 * c] / (float)M;
  float var = stats[2 * c + 1] / (float)M - mean * mean;
  float inv = rsqrtf(var + EPSV);
  float v = (Y[(size_t)r * ldy + c] - mean) * inv * g[c] + b[c];
  if (resid) v += resid[(size_t)r * ldr + c];
  if (relu) v = fmaxf(v, 0.f);
  O[(size_t)r * ldo + c] = v;
}

// ---------------------------------------------------------------------------
// KNN top-32 by pd = -||xi-xj||^2, tie-broken by smaller index (top_k order).
// One wave per query point; distance row cached in LDS; 32 selection passes.
// ---------------------------------------------------------------------------
__global__ __launch_bounds__(256) void knn_topk(const float* __restrict__ x,
                                                int* __restrict__ knn) {
  __shared__ float dist[8][NPTS];  // 64 KB
  int wave = threadIdx.x >> 5, lane = threadIdx.x & 31;
  int point = blockIdx.x * 8 + wave;     // < 32768 always (4096 blocks)
  int b = point >> 11, i = point & (NPTS - 1);
  const float* xb = x + (size_t)b * 3 * NPTS;
  float xi0 = xb[i], xi1 = xb[NPTS + i], xi2 = xb[2 * NPTS + i];
  float* drow = dist[wave];
  for (int j = lane; j < NPTS; j += 32) {
    float d0 = xi0 - xb[j];
    float d1 = xi1 - xb[NPTS + j];
    float d2 = xi2 - xb[2 * NPTS + j];
    drow[j] = -(d0 * d0 + d1 * d1 + d2 * d2);
  }
  __syncthreads();
  for (int s = 0; s < KNN; ++s) {
    float best = -3.4e38f;
    int bidx = NPTS;
    for (int j = lane; j < NPTS; j += 32) {
      float v = drow[j];
      if (v > best || (v == best && j < bidx)) { best = v; bidx = j; }
    }
    for (int off = 16; off; off >>= 1) {
      float ov = __shfl_xor(best, off, 32);
      int oi = __shfl_xor(bidx, off, 32);
      if (ov > best || (ov == best && oi < bidx)) { best = ov; bidx = oi; }
    }
    if (lane == 0) knn[(size_t)point * KNN + s] = bidx;
    if (lane == (bidx & 31)) drow[bidx] = -3.4e38f;  // remove winner
    __syncthreads();
  }
}

// ---------------------------------------------------------------------------
// GeometricAffine sigma = mean over (B,N,K,C) of offset^2 (deterministic 2-stage)
// ---------------------------------------------------------------------------
__global__ __launch_bounds__(256) void ga_sigma_partial(
    const float* __restrict__ F, int ld, const int* __restrict__ knn,
    float* __restrict__ partial) {
  __shared__ float ssum[256];
  const int NT = MROWS * KNN;  // 1,048,576 (point,k) pairs
  float acc = 0.f;
  for (int t = blockIdx.x * 256 + threadIdx.x; t < NT; t += 1024 * 256) {
    int p = t >> 5;
    int j = knn[t];
    int b = p >> 11;
    const float* fi = F + (size_t)p * ld;
    const float* fj = F + (size_t)(b * NPTS + j) * ld;
#pragma unroll
    for (int c = 0; c < 8; ++c) {
      float d = fj[c] - fi[c];
      acc += d * d;
    }
  }
  ssum[threadIdx.x] = acc;
  __syncthreads();
  for (int o = 128; o; o >>= 1) {
    if (threadIdx.x < o) ssum[threadIdx.x] += ssum[threadIdx.x + o];
    __syncthreads();
  }
  if (threadIdx.x == 0) partial[blockIdx.x] = ssum[0];
}

__global__ __launch_bounds__(256) void ga_sigma_reduce(
    const float* __restrict__ partial, float* __restrict__ sigma) {
  __shared__ float s[256];
  float a = partial[threadIdx.x] + partial[threadIdx.x + 256] +
            partial[threadIdx.x + 512] + partial[threadIdx.x + 768];
  s[threadIdx.x] = a;
  __syncthreads();
  for (int o = 128; o; o >>= 1) {
    if (threadIdx.x < o) s[threadIdx.x] += s[threadIdx.x + o];
    __syncthreads();
  }
  if (threadIdx.x == 0)
    sigma[0] = s[0] / ((float)MROWS * (float)KNN * 8.f);
}

__global__ void ga_apply(const float* __restrict__ F, int ld,
                         const int* __restrict__ knn,
                         const float* __restrict__ sigma,
                         const float* __restrict__ alpha,
                         const float* __restrict__ beta,
                         float* __restrict__ O, int ldo) {
  int i = blockIdx.x * 256 + threadIdx.x;  // point*8 + c
  if (i >= MROWS * 8) return;
  int p = i >> 3, c = i & 7;
  int b = p >> 11;
  float inv = 1.f / (sigma[0] + EPSV);
  float fi = F[(size_t)p * ld + c];
  float al = alpha[c], be = beta[c];
  float mx = -3.4e38f;
  const int* kr = knn + (size_t)p * KNN;
  for (int k = 0; k < KNN; ++k) {
    int j = kr[k];
    float off = F[(size_t)(b * NPTS + j) * ld + c] - fi;
    mx = fmaxf(mx, off * inv * al + be);
  }
  O[(size_t)p * ldo + c] = mx;
}

// ---------------------------------------------------------------------------
// Per-(b,c) max & mean over N points. One block per (b,c), LDS tree reduce.
// ---------------------------------------------------------------------------
__global__ __launch_bounds__(256) void pool_maxmean(const float* __restrict__ X,
                                                    int ld, int C,
                                                    float* __restrict__ pmax,
                                                    float* __restrict__ pmean) {
  __shared__ float sm[256], ss[256];
  int b = blockIdx.x / C, c = blockIdx.x - b * C;
  const float* base = X + (size_t)b * NPTS * ld + c;
  float mx = -3.4e38f, s = 0.f;
  for (int n = threadIdx.x; n < NPTS; n += 256) {
    float v = base[(size_t)n * ld];
    mx = fmaxf(mx, v);
    s += v;
  }
  sm[threadIdx.x] = mx;
  ss[threadIdx.x] = s;
  __syncthreads();
  for (int o = 128; o; o >>= 1) {
    if (threadIdx.x < o) {
      sm[threadIdx.x] = fmaxf(sm[threadIdx.x], sm[threadIdx.x + o]);
      ss[threadIdx.x] += ss[threadIdx.x + o];
    }
    __syncthreads();
  }
  if (threadIdx.x == 0) {
    pmax[blockIdx.x] = sm[0];
    pmean[blockIdx.x] = ss[0] * (1.f / (float)NPTS);
  }
}

// ---------------------------------------------------------------------------
// Build concat [x | g_max | g_mean] (row stride 96)
// ---------------------------------------------------------------------------
__global__ void build_concat(const float* __restrict__ X, int ldx,
                             const float* __restrict__ pmax,
                             const float* __restrict__ pmean,
                             float* __restrict__ Hc, int C) {
  int i = blockIdx.x * 256 + threadIdx.x;
  if (i >= MROWS * C) return;
  int r = i / C, c = i - r * C;
  int b = r >> 11;
  float* h = Hc + (size_t)r * 96;
  h[c] = X[(size_t)r * ldx + c];
  h[C + c] = pmax[b * C + c];
  h[2 * C + c] = pmean[b * C + c];
}

// ---------------------------------------------------------------------------
// Head: 16x32 pooled -> fc1(16)+BN+relu -> fc2(8)+b+BN+relu -> fc3(40)+b
// Single block, all in LDS, deterministic.
// ---------------------------------------------------------------------------
__global__ __launch_bounds__(256) void head_kernel(
    const float* __restrict__ pooled, const float* __restrict__ hfc1,
    const float* __restrict__ g1, const float* __restrict__ b1,
    const float* __restrict__ hfc2, const float* __restrict__ hb2,
    const float* __restrict__ g2, const float* __restrict__ bb2,
    const float* __restrict__ hfc3, const float* __restrict__ hb3,
    float* __restrict__ out) {
  __shared__ float F[16 * 32], Y1[16 * 16], Y2[16 * 8], mstat[32];
  int tid = threadIdx.x;
  for (int i = tid; i < 16 * 32; i += 256) F[i] = pooled[i];
  __syncthreads();
  {  // y1 = F @ hfc1 (16x16); one element per thread
    int r = tid >> 4, c = tid & 15;
    float a = 0.f;
    for (int k = 0; k < 32; ++k) a += F[r * 32 + k] * hfc1[k * 16 + c];
    Y1[r * 16 + c] = a;
  }
  __syncthreads();
  if (tid < 16) {
    float s = 0.f, q = 0.f;
    for (int r = 0; r < 16; ++r) {
      float v = Y1[r * 16 + tid];
      s += v;
      q += v * v;
    }
    float m = s / 16.f, var = q / 16.f - m * m;
    mstat[tid] = m;
    mstat[16 + tid] = rsqrtf(var + EPSV);
  }
  __syncthreads();
  {
    int r = tid >> 4, c = tid & 15;
    float v = (Y1[r * 16 + c] - mstat[c]) * mstat[16 + c] * g1[c] + b1[c];
    Y1[r * 16 + c] = fmaxf(v, 0.f);
  }
  __syncthreads();
  if (tid < 128) {  // y2 = Y1 @ hfc2 + hb2 (16x8)
    int r = tid >> 3, c = tid & 7;
    float a = hb2[c];
    for (int k = 0; k < 16; ++k) a += Y1[r * 16 + k] * hfc2[k * 8 + c];
    Y2[r * 8 + c] = a;
  }
  __syncthreads();
  if (tid < 8) {
    float s = 0.f, q = 0.f;
    for (int r = 0; r < 16; ++r) {
      float v = Y2[r * 8 + tid];
      s += v;
      q += v * v;
    }
    float m = s / 16.f, var = q / 16.f - m * m;
    mstat[tid] = m;
    mstat[16 + tid] = rsqrtf(var + EPSV);
  }
  __syncthreads();
  if (tid < 128) {
    int r = tid >> 3, c = tid & 7;
    float v = (Y2[r * 8 + c] - mstat[c]) * mstat[16 + c] * g2[c] + bb2[c];
    Y2[r * 8 + c] = fmaxf(v, 0.f);
  }
  __syncthreads();
  for (int e = tid; e < 16 * 40; e += 256) {  // y3 = Y2 @ hfc3 + hb3
    int r = e / 40, c = e - r * 40;
    float a = hb3[c];
    for (int k = 0; k < 8; ++k) a += Y2[r * 8 + k] * hfc3[k * 40 + c];
    out[e] = a;
  }
}

// ---------------------------------------------------------------------------
// Host orchestration
// ---------------------------------------------------------------------------
static float* g_wt = nullptr;  // never assigned; unused (no state kept)

static inline void launch_gemm(const float* A, int lda, const float* W,
                               float* WT, float* Y, int ldy, int K, int Nc,
                               hipStream_t s, int M = MROWS) {
  int ntn = (Nc + 15) / 16;
  int npad = ntn * 16;
  transpose_pad_w<<<(npad * K + 255) / 256, 256, 0, s>>>(W, WT, K, Nc, npad);
  int tiles = (M / 16) * ntn;
  int grid = (tiles + 7) / 8;
  gemm_wmma_f32<<<grid, 256, 0, s>>>(A, WT, Y, K, Nc, lda, ldy, ntn, tiles);
}

extern "C" void kernel_launch(void* const* d_in, const int* in_sizes, int n_in,
                              void* d_out, int out_size, void* d_ws,
                              size_t ws_size, hipStream_t stream) {
  (void)in_sizes; (void)n_in; (void)out_size; (void)ws_size; (void)g_wt;
  const float* x = (const float*)d_in[0];
  auto f = [&](int i) { return (const float*)d_in[i]; };

  // workspace layout (floats)
  float* ws = (float*)d_ws;
  size_t o = 0;
  float* P0 = ws + o;  o += (size_t)MROWS * 64;
  float* P1 = ws + o;  o += (size_t)MROWS * 64;
  float* Yb = ws + o;  o += (size_t)MROWS * 64;
  float* Hb = ws + o;  o += (size_t)MROWS * 96;
  float* IDb = ws + o; o += (size_t)MROWS * 32;
  int* knn = (int*)(ws + o); o += (size_t)MROWS * KNN;
  float* stats = ws + o;    o += 128;
  float* pmax = ws + o;     o += BQ * 64;
  float* pmean = ws + o;    o += BQ * 64;
  float* partials = ws + o; o += 1024;
  float* sigma = ws + o;    o += 4;
  float* pooled = ws + o;   o += BQ * 32;
  float* WT = ws + o;       o += 64 * 64;  // transposed/padded weight scratch

  // ---- stem: relu(BN(x^T @ stem_fc)) -> P0 (C=8) ----
  stem_kernel<<<MROWS / 256, 256, 0, stream>>>(x, f(1), Yb, 64);
  bn_stats<<<8, 256, 0, stream>>>(Yb, MROWS, 64, stats);
  bn_apply<<<(MROWS * 8 + 255) / 256, 256, 0, stream>>>(
      Yb, 64, P0, 64, stats, f(2), f(3), nullptr, 0, MROWS, 8, 1);

  // ---- KNN + GeometricAffine -> P1 (C=8) ----
  knn_topk<<<MROWS / 8, 256, 0, stream>>>(x, knn);
  ga_sigma_partial<<<1024, 256, 0, stream>>>(P0, 64, knn, partials);
  ga_sigma_reduce<<<1, 256, 0, stream>>>(partials, sigma);
  ga_apply<<<(MROWS * 8 + 255) / 256, 256, 0, stream>>>(P0, 64, knn, sigma,
                                                        f(4), f(5), P1, 64);

  float* cur = P1;
  float* oth = P0;

  // rmb helper: in(ld) --fc1/bn/relu--> H --fc2/bn(+resid)/relu--> out(ld)
  auto run_rmb = [&](const float* in, int ldin, float* outp, int ldout,
                     int base, int Cin, int Cout, int hid, int projIdx) {
    launch_gemm(in, ldin, f(base + 0), WT, Yb, 64, Cin, hid, stream);
    bn_stats<<<hid, 256, 0, stream>>>(Yb, MROWS, 64, stats);
    bn_apply<<<(MROWS * hid + 255) / 256, 256, 0, stream>>>(
        Yb, 64, Hb, 96, stats, f(base + 1), f(base + 2), nullptr, 0, MROWS,
        hid, 1);
    launch_gemm(Hb, 96, f(base + 3), WT, Yb, 64, hid, Cout, stream);
    bn_stats<<<Cout, 256, 0, stream>>>(Yb, MROWS, 64, stats);
    const float* resid = in;
    int ldr = ldin;
    if (projIdx >= 0) {
      launch_gemm(in, ldin, f(projIdx), WT, IDb, 32, Cin, Cout, stream);
      resid = IDb;
      ldr = 32;
    }
    bn_apply<<<(MROWS * Cout + 255) / 256, 256, 0, stream>>>(
        Yb, 64, outp, ldout, stats, f(base + 4), f(base + 5), resid, ldr,
        MROWS, Cout, 1);
  };

  auto run_fuse = [&](const float* in, int ldin, float* outp, int ldout,
                      int base, int C) {
    pool_maxmean<<<BQ * C, 256, 0, stream>>>(in, ldin, C, pmax, pmean);
    build_concat<<<(MROWS * C + 255) / 256, 256, 0, stream>>>(in, ldin, pmax,
                                                              pmean, Hb, C);
    launch_gemm(Hb, 96, f(base + 0), WT, Yb, 64, 3 * C, C, stream);
    bn_stats<<<C, 256, 0, stream>>>(Yb, MROWS, 64, stats);
    bn_apply<<<(MROWS * C + 255) / 256, 256, 0, stream>>>(
        Yb, 64, outp, ldout, stats, f(base + 1), f(base + 2), nullptr, 0,
        MROWS, C, 1);
  };

  struct StageCfg { int pre, fuse, post, proj, Cin, Cout, hpre, hpost; };
  const StageCfg stages[3] = {
      {6, 12, 15, -1, 8, 8, 4, 32},    // s1
      {21, 27, 30, 36, 8, 16, 4, 32},  // s2 (post has proj @ idx 36)
      {37, 43, 46, 52, 16, 32, 8, 64}, // s3 (post has proj @ idx 52)
  };

  for (int s = 0; s < 3; ++s) {
    const StageCfg& c = stages[s];
    run_rmb(cur, 64, oth, 64, c.pre, c.Cin, c.Cin, c.hpre, -1);  // pre
    run_fuse(oth, 64, cur, 64, c.fuse, c.Cin);                   // fuse
    run_rmb(cur, 64, oth, 64, c.post, c.Cin, c.Cout, c.hpost, c.proj); // post
    float* t = cur; cur = oth; oth = t;
  }

  // ---- global max pool over N, then head ----
  pool_maxmean<<<BQ * 32, 256, 0, stream>>>(cur, 64, 32, pooled, pmean);
  head_kernel<<<1, 256, 0, stream>>>(pooled, f(53), f(54), f(55), f(56), f(57),
                                     f(58), f(59), f(60), f(61),
                                     (float*)d_out);
}